// GAT_64149631533290
// MI455X (gfx1250) — compile-verified
//
#include <hip/hip_runtime.h>
#include <hip/hip_bf16.h>
#include <math.h>

typedef float v2f __attribute__((ext_vector_type(2)));
typedef float v8f __attribute__((ext_vector_type(8)));

#define NEG_SLOPE 0.2f

// ============================================================================
// GEMM via V_WMMA_F32_16X16X4_F32:  H[n,fout] = X[n,fin] @ W[fin,fout]
// Preconditions: n % 16 == 0, fin % 4 == 0, fout % 16 == 0 (all hold here).
// 128-thread block = 4 waves; each wave computes one 16x16 tile of H.
// Guard is wave-uniform, so EXEC is all-ones around every WMMA.
// ============================================================================
__global__ __launch_bounds__(128)
void gat_gemm_wmma(const float* __restrict__ X, const float* __restrict__ W,
                   float* __restrict__ H, int n, int fin, int fout) {
    const int wave = threadIdx.x >> 5;            // 0..3
    const int tm   = blockIdx.x * 4 + wave;       // row tile for this wave
    if (tm * 16 >= n) return;                     // wave-uniform exit
    const int tn   = blockIdx.y;                  // col tile
    const int lane = threadIdx.x & 31;
    const int half = lane >> 4;                   // 0: K=k0,k0+1 ; 1: K=k0+2,k0+3
    const int l16  = lane & 15;
    const int row  = tm * 16 + l16;               // A-matrix row held by this lane
    const int col  = tn * 16 + l16;               // B/C/D column held by this lane

    v8f acc = {};
    for (int k0 = 0; k0 < fin; k0 += 4) {
        const int ka = k0 + half * 2;
        // A 16x4 fp32: lane holds A[row][ka], A[row][ka+1] (8B-aligned pair)
        v2f a = *(const v2f*)(X + (size_t)row * fin + ka);
        // B 4x16 fp32: VGPR0 = W[ka][col], VGPR1 = W[ka+1][col]
        const float* wp = W + (size_t)ka * fout + col;
        v2f b;
        b.x = wp[0];
        b.y = wp[fout];
        acc = __builtin_amdgcn_wmma_f32_16x16x4_f32(false, a, false, b,
                                                    (short)0, acc, false, false);
    }
    // C/D 16x16 fp32: VGPR r -> row (half*8 + r), column = col
    const int rbase = tm * 16 + half * 8;
    #pragma unroll
    for (int r = 0; r < 8; ++r)
        H[(size_t)(rbase + r) * fout + col] = acc[r];
}

// ============================================================================
// Per-node attention scores: ssrc[i] = h_i . a_s ; sdst[i] = h_i . a_d
// ============================================================================
__global__ __launch_bounds__(256)
void gat_scores(const float* __restrict__ H, const float* __restrict__ as,
                const float* __restrict__ ad, float* __restrict__ ssrc,
                float* __restrict__ sdst, int n, int f) {
    int i = blockIdx.x * blockDim.x + threadIdx.x;
    if (i >= n) return;
    const float4* hp = (const float4*)(H + (size_t)i * f);
    const float4* ap = (const float4*)as;
    const float4* dp = (const float4*)ad;
    float s1 = 0.f, s2 = 0.f;
    for (int j = 0; j < (f >> 2); ++j) {
        float4 h = hp[j], a = ap[j], d = dp[j];
        s1 = fmaf(h.x, a.x, fmaf(h.y, a.y, fmaf(h.z, a.z, fmaf(h.w, a.w, s1))));
        s2 = fmaf(h.x, d.x, fmaf(h.y, d.y, fmaf(h.z, d.z, fmaf(h.w, d.w, s2))));
    }
    ssrc[i] = s1;
    sdst[i] = s2;
}

// ---- order-preserving float <-> uint key (for atomic max on floats) --------
__device__ __forceinline__ unsigned f2key(float f) {
    unsigned u = __float_as_uint(f);
    return (u & 0x80000000u) ? ~u : (u | 0x80000000u);
}
__device__ __forceinline__ float key2f(unsigned u) {
    return (u & 0x80000000u) ? __uint_as_float(u ^ 0x80000000u)
                             : __uint_as_float(~u);
}

// ============================================================================
// Edge pass 1: logits = leaky_relu(ssrc[src]+sdst[dst]); segment max via
// native global_atomic_max_u32 on monotone keys. Edges e >= E are self-loops.
// ============================================================================
__global__ __launch_bounds__(256)
void gat_edge_logits_max(const int* __restrict__ src, const int* __restrict__ dst,
                         int E, int n,
                         const float* __restrict__ ssrc, const float* __restrict__ sdst,
                         float* __restrict__ elog, unsigned* __restrict__ mkey) {
    int e = blockIdx.x * blockDim.x + threadIdx.x;
    int Et = E + n;
    if (e >= Et) return;
    int s, d;
    if (e < E) { s = src[e]; d = dst[e]; } else { s = d = e - E; }
    float l = ssrc[s] + sdst[d];
    l = (l > 0.f) ? l : NEG_SLOPE * l;
    elog[e] = l;
    atomicMax(mkey + d, f2key(l));
}

// ============================================================================
// Edge pass 2: w = exp(logit - m[dst]); denom[dst] += w  (w stored in place)
// ============================================================================
__global__ __launch_bounds__(256)
void gat_edge_exp_sum(const int* __restrict__ dst, int E, int n,
                      const unsigned* __restrict__ mkey,
                      float* __restrict__ ew, float* __restrict__ denom) {
    int e = blockIdx.x * blockDim.x + threadIdx.x;
    int Et = E + n;
    if (e >= Et) return;
    int d = (e < E) ? dst[e] : (e - E);
    float m = key2f(mkey[d]);
    if (!isfinite(m)) m = 0.f;
    float w = __expf(ew[e] - m);
    ew[e] = w;
    atomicAdd(denom + d, w);
}

// ============================================================================
// Edge pass 3: accum[dst][:] += w_e * h[src][:]   (division by denom deferred)
// One thread per (edge, 4-feature chunk): f/4 consecutive lanes cover one
// edge, so the h[src] gather is a coalesced burst and the atomic adds hit
// consecutive addresses within the wave.
// ============================================================================
__global__ __launch_bounds__(256)
void gat_edge_aggregate(const int* __restrict__ src, const int* __restrict__ dst,
                        int E, int n, int f, int cshift /* log2(f/4) */,
                        const float* __restrict__ ew, const float* __restrict__ H,
                        float* __restrict__ accum) {
    int gid = blockIdx.x * blockDim.x + threadIdx.x;
    int e   = gid >> cshift;
    int Et  = E + n;
    if (e >= Et) return;
    int c = (gid & ((1 << cshift) - 1)) << 2;     // feature offset (0,4,8,...)
    int s, d;
    if (e < E) { s = src[e]; d = dst[e]; } else { s = d = e - E; }
    float  w  = ew[e];
    float4 hv = *(const float4*)(H + (size_t)s * f + c);
    float* ap = accum + (size_t)d * f + c;
    atomicAdd(ap + 0, w * hv.x);
    atomicAdd(ap + 1, w * hv.y);
    atomicAdd(ap + 2, w * hv.z);
    atomicAdd(ap + 3, w * hv.w);
}

// ============================================================================
// Finalize (layers 1,2): out = relu(accum/denom + b)   (float4 per thread)
// ============================================================================
__global__ __launch_bounds__(256)
void gat_finalize_relu(const float* __restrict__ accum, const float* __restrict__ denom,
                       const float* __restrict__ b, float* __restrict__ out,
                       int n, int f, int cshift /* log2(f/4) */) {
    int idx = blockIdx.x * blockDim.x + threadIdx.x;     // one float4 per thread
    if (idx >= (n << cshift)) return;
    int i = idx >> cshift;
    int j = (idx & ((1 << cshift) - 1)) << 2;
    float inv = 1.f / (denom[i] + 1e-16f);
    float4 a  = *(const float4*)(accum + ((size_t)idx << 2));
    float4 bb = *(const float4*)(b + j);
    float4 v;
    v.x = fmaxf(a.x * inv + bb.x, 0.f);
    v.y = fmaxf(a.y * inv + bb.y, 0.f);
    v.z = fmaxf(a.z * inv + bb.z, 0.f);
    v.w = fmaxf(a.w * inv + bb.w, 0.f);
    *(float4*)(out + ((size_t)idx << 2)) = v;
}

// ============================================================================
// Finalize (layer 3): out = log_softmax(accum/denom + b) over 32 features
// ============================================================================
__global__ __launch_bounds__(256)
void gat_finalize_logsoftmax(const float* __restrict__ accum,
                             const float* __restrict__ denom,
                             const float* __restrict__ b, float* __restrict__ out,
                             int n) {
    int i = blockIdx.x * blockDim.x + threadIdx.x;
    if (i >= n) return;
    float inv = 1.f / (denom[i] + 1e-16f);
    float v[32];
    float mx = -INFINITY;
    #pragma unroll
    for (int j = 0; j < 32; ++j) {
        float t = accum[(size_t)i * 32 + j] * inv + b[j];
        v[j] = t;
        mx = fmaxf(mx, t);
    }
    float s = 0.f;
    #pragma unroll
    for (int j = 0; j < 32; ++j) s += __expf(v[j] - mx);
    float lse = mx + __logf(s);
    #pragma unroll
    for (int j = 0; j < 32; ++j) out[(size_t)i * 32 + j] = v[j] - lse;
}

// ============================================================================
// Host-side layer driver (all work queued on `stream`; graph-capture safe)
// ============================================================================
static void run_gat_layer(const float* X, int fin, int fout,
                          const float* W, const float* as, const float* ad,
                          const float* bias,
                          const int* src, const int* dst, int E, int n,
                          float* H, float* accum, float* ssrc, float* sdst,
                          float* denom, unsigned* mkey, float* ew,
                          float* out, bool last, hipStream_t stream) {
    const int Et     = E + n;
    const int cshift = (fout == 64) ? 4 : 3;      // log2(fout/4)
    // h = X @ W  (WMMA tiles; 4 waves / block)
    const int mtiles = n / 16;
    gat_gemm_wmma<<<dim3((mtiles + 3) / 4, fout / 16), 128, 0, stream>>>(
        X, W, H, n, fin, fout);
    // per-node scores
    gat_scores<<<(n + 255) / 256, 256, 0, stream>>>(H, as, ad, ssrc, sdst, n, fout);
    // reset segment buffers
    hipMemsetAsync(mkey, 0, (size_t)n * sizeof(unsigned), stream);
    hipMemsetAsync(denom, 0, (size_t)n * sizeof(float), stream);
    hipMemsetAsync(accum, 0, (size_t)n * fout * sizeof(float), stream);
    // edge passes
    gat_edge_logits_max<<<(Et + 255) / 256, 256, 0, stream>>>(src, dst, E, n,
                                                              ssrc, sdst, ew, mkey);
    gat_edge_exp_sum<<<(Et + 255) / 256, 256, 0, stream>>>(dst, E, n, mkey, ew, denom);
    {
        long work = (long)Et << cshift;
        gat_edge_aggregate<<<(int)((work + 255) / 256), 256, 0, stream>>>(
            src, dst, E, n, fout, cshift, ew, H, accum);
    }
    // finalize
    if (last) {
        gat_finalize_logsoftmax<<<(n + 255) / 256, 256, 0, stream>>>(accum, denom,
                                                                     bias, out, n);
    } else {
        long work = (long)n << cshift;
        gat_finalize_relu<<<(int)((work + 255) / 256), 256, 0, stream>>>(
            accum, denom, bias, out, n, fout, cshift);
    }
}

extern "C" void kernel_launch(void* const* d_in, const int* in_sizes, int n_in,
                              void* d_out, int out_size, void* d_ws, size_t ws_size,
                              hipStream_t stream) {
    const float* x   = (const float*)d_in[0];
    const int*   ei  = (const int*)d_in[1];
    const float* W1  = (const float*)d_in[2];
    const float* a1s = (const float*)d_in[3];
    const float* a1d = (const float*)d_in[4];
    const float* b1  = (const float*)d_in[5];
    const float* W2  = (const float*)d_in[6];
    const float* a2s = (const float*)d_in[7];
    const float* a2d = (const float*)d_in[8];
    const float* b2  = (const float*)d_in[9];
    const float* W3  = (const float*)d_in[10];
    const float* a3s = (const float*)d_in[11];
    const float* a3d = (const float*)d_in[12];
    const float* b3  = (const float*)d_in[13];

    const int n = in_sizes[0] / 128;   // 100000
    const int E = in_sizes[1] / 2;     // 1600000
    const int* src = ei;               // edge_index[0]
    const int* dst = ei + E;           // edge_index[1]

    // ---- workspace carve-out (floats) ----
    float* base  = (float*)d_ws;
    float* hA    = base;                       // n*64
    float* hB    = hA + (size_t)n * 64;        // n*64
    float* accum = hB + (size_t)n * 64;        // n*64
    float* ssrc  = accum + (size_t)n * 64;     // n
    float* sdst  = ssrc + n;                   // n
    float* denom = sdst + n;                   // n
    unsigned* mkey = (unsigned*)(denom + n);   // n
    float* ew    = (float*)(mkey + n);         // E + n

    // Layer 1: x[n,128] -> hB[n,64] (relu)
    run_gat_layer(x, 128, 64, W1, a1s, a1d, b1, src, dst, E, n,
                  hA, accum, ssrc, sdst, denom, mkey, ew, hB, false, stream);
    // Layer 2: GEMM reads hB -> hA; aggregate reads hA; finalize -> hB
    run_gat_layer(hB, 64, 64, W2, a2s, a2d, b2, src, dst, E, n,
                  hA, accum, ssrc, sdst, denom, mkey, ew, hB, false, stream);
    // Layer 3: hB[n,64] -> d_out[n,32] (log_softmax)
    run_gat_layer(hB, 64, 32, W3, a3s, a3d, b3, src, dst, E, n,
                  hA, accum, ssrc, sdst, denom, mkey, ew, (float*)d_out, true, stream);
}